// Model_2_82343112999185
// MI455X (gfx1250) — compile-verified
//
#include <hip/hip_runtime.h>
#include <hip/hip_bf16.h>
#include <math.h>

typedef __attribute__((ext_vector_type(2))) float v2f;
typedef __attribute__((ext_vector_type(8))) float v8f;

#define SEQ   1024
#define INDIM 400
#define Hdim  512
#define H4    2048
#define Tdim  13

__device__ __forceinline__ float sigmf(float x) { return 1.0f / (1.0f + __expf(-x)); }

// ---------------- Kernel A: embedding gather -> feat [SEQ, 400] ----------------
__global__ void gather_kernel(const int* __restrict__ x, const int* __restrict__ cas,
                              const int* __restrict__ pos,
                              const float* __restrict__ W_emb, const float* __restrict__ W_cas,
                              const float* __restrict__ W_pos, float* __restrict__ feat) {
    int t = blockIdx.x;
    int xi = x[t], ci = cas[t], pi = pos[t];
    for (int d = threadIdx.x; d < INDIM; d += blockDim.x) {
        float v;
        if (d < 300)       v = W_emb[(size_t)xi * 300 + d];
        else if (d < 350)  v = W_cas[ci * 50 + (d - 300)];
        else               v = W_pos[pi * 50 + (d - 350)];
        feat[(size_t)t * INDIM + d] = v;
    }
}

// ---------------- Kernel B: WMMA f32 GEMM  pre[dir] = feat @ Wx_dir + b_dir ----------------
// Tiles: dir(2) x Mtiles(64) x Ntiles(128); one 16x16 tile per wave, K stepped by 4.
__global__ __launch_bounds__(256) void preproj_wmma_kernel(
    const float* __restrict__ feat,
    const float* __restrict__ Wx_f, const float* __restrict__ b_f,
    const float* __restrict__ Wx_b, const float* __restrict__ b_b,
    float* __restrict__ pre) {
    const int wid  = threadIdx.x >> 5;          // wave in block (8 waves)
    const int lane = threadIdx.x & 31;
    const int tile = blockIdx.x * 8 + wid;      // 0..16383
    const int dir  = tile / (64 * 128);
    const int rem  = tile % (64 * 128);
    const int mt   = rem / 128;
    const int nt   = rem % 128;

    const int r  = lane & 15;   // M row for A, N col for B/D
    const int hi = lane >> 4;   // lane-half selects K pair

    const float* __restrict__ Wx = dir ? Wx_b : Wx_f;
    const float* __restrict__ bv = dir ? b_b  : b_f;

    const float* __restrict__ Arow = feat + (size_t)(mt * 16 + r) * INDIM;
    const float* __restrict__ Bcol = Wx + nt * 16 + r;

    v8f acc = {};
    #pragma unroll 4
    for (int k = 0; k < INDIM; k += 4) {
        const int ka = k + 2 * hi;
        v2f a, b;
        a.x = Arow[ka];
        a.y = Arow[ka + 1];
        b.x = Bcol[(size_t)ka * H4];
        b.y = Bcol[(size_t)(ka + 1) * H4];
        acc = __builtin_amdgcn_wmma_f32_16x16x4_f32(
            /*neg_a=*/false, a, /*neg_b=*/false, b,
            /*c_mod=*/(short)0, acc, /*reuse_a=*/false, /*reuse_b=*/false);
    }

    const float bias = bv[nt * 16 + r];
    float* __restrict__ out = pre + ((size_t)dir * SEQ + (size_t)mt * 16) * H4 + nt * 16;
    #pragma unroll
    for (int v = 0; v < 8; ++v) {
        out[(size_t)(v + 8 * hi) * H4 + r] = acc[v] + bias;
    }
}

// ---------------- Kernel C: bi-LSTM recurrence, Wh slices resident in 320KB LDS ----------------
// 32 blocks x 128 threads. blockIdx: dir = b>>4, wg = b&15. Each WG owns 32 hidden dims
// (j0 = wg*32) and the 128 z-columns {g*512 + j0 + idx : g in 0..3}. Wh slice 512x128 = 256KB LDS.
__global__ __launch_bounds__(128) void lstm_kernel(
    const float* __restrict__ pre,
    const float* __restrict__ Wh_f, const float* __restrict__ Wh_b,
    const float* __restrict__ h0,   const float* __restrict__ c0,
    float* __restrict__ hs, float* __restrict__ Hbuf, int* __restrict__ cnt) {
    extern __shared__ float smem[];
    float* Whs  = smem;                 // 512*128
    float* hld  = Whs + Hdim * 128;     // 512
    float* zbuf = hld + Hdim;           // 128
    float* cbuf = zbuf + 128;           // 32

    const int tid = threadIdx.x;
    const int dir = blockIdx.x >> 4;
    const int wg  = blockIdx.x & 15;
    const int j0  = wg * 32;
    const int c   = tid;                // owned z-column index, 0..127
    const int gate = c >> 5;
    const int idx  = c & 31;
    const int J    = gate * Hdim + j0 + idx;   // global z column

    const float* __restrict__ Wh = dir ? Wh_b : Wh_f;
    // stage Wh slice into LDS
    for (int k = 0; k < Hdim; ++k) Whs[k * 128 + c] = Wh[(size_t)k * H4 + J];
    for (int k = tid; k < Hdim; k += 128) hld[k] = h0[dir * Hdim + k];
    if (tid < 32) cbuf[tid] = c0[dir * Hdim + j0 + tid];
    __syncthreads();

    for (int s = 0; s < SEQ; ++s) {
        const int t = dir ? (SEQ - 1 - s) : s;
        const float* __restrict__ prow = pre + ((size_t)dir * SEQ + t) * H4;
        float acc = prow[J];
        #pragma unroll 8
        for (int k = 0; k < Hdim; ++k) acc = fmaf(hld[k], Whs[k * 128 + c], acc);
        zbuf[c] = acc;
        __syncthreads();

        if (tid < 32) {
            const float iv = zbuf[tid], fv = zbuf[32 + tid];
            const float gv = zbuf[64 + tid], ov = zbuf[96 + tid];
            const float cc = sigmf(fv) * cbuf[tid] + sigmf(iv) * tanhf(gv);
            cbuf[tid] = cc;
            const float hh = sigmf(ov) * tanhf(cc);
            const int j = j0 + tid;
            hs[((size_t)dir * SEQ + t) * Hdim + j] = hh;
            Hbuf[(dir * 2 + (s & 1)) * Hdim + j]  = hh;   // ping-pong exchange buffer
        }
        __syncthreads();

        // cross-workgroup barrier (per direction), monotonic counter zeroed per launch
        if (tid == 0) {
            __threadfence();
            atomicAdd(&cnt[dir], 1);
            const int target = 16 * (s + 1);
            while (__hip_atomic_load(&cnt[dir], __ATOMIC_RELAXED, __HIP_MEMORY_SCOPE_AGENT) < target) {
                __builtin_amdgcn_s_sleep(1);
            }
            __threadfence();
        }
        __syncthreads();

        for (int k = tid; k < Hdim; k += 128)
            hld[k] = Hbuf[(dir * 2 + (s & 1)) * Hdim + k];
        __syncthreads();
    }
}

// ---------------- Kernel D: output projection + log_softmax ----------------
__global__ __launch_bounds__(64) void proj_softmax_kernel(
    const float* __restrict__ hs, const float* __restrict__ W_out,
    const float* __restrict__ b_out, float* __restrict__ tag_scores) {
    __shared__ float hvec[2 * Hdim];
    __shared__ float obuf[Tdim];
    __shared__ float red[2];
    const int t = blockIdx.x, tid = threadIdx.x;
    for (int m = tid; m < 2 * Hdim; m += blockDim.x) {
        hvec[m] = (m < Hdim) ? hs[((size_t)0 * SEQ + t) * Hdim + m]
                             : hs[((size_t)1 * SEQ + t) * Hdim + (m - Hdim)];
    }
    __syncthreads();
    if (tid < Tdim) {
        float a = b_out[tid];
        for (int m = 0; m < 2 * Hdim; ++m) a = fmaf(hvec[m], W_out[m * Tdim + tid], a);
        obuf[tid] = a;
    }
    __syncthreads();
    if (tid == 0) {
        float mx = obuf[0];
        for (int j = 1; j < Tdim; ++j) mx = fmaxf(mx, obuf[j]);
        float sm = 0.0f;
        for (int j = 0; j < Tdim; ++j) sm += __expf(obuf[j] - mx);
        red[0] = mx; red[1] = __logf(sm);
    }
    __syncthreads();
    if (tid < Tdim) tag_scores[(size_t)t * Tdim + tid] = obuf[tid] - red[0] - red[1];
}

// ---------------- Kernel E: Viterbi (max-plus DP over T=13) ----------------
__global__ __launch_bounds__(32) void viterbi_kernel(
    const float* __restrict__ em, const float* __restrict__ trans,
    const float* __restrict__ start_t, const float* __restrict__ end_t,
    int* __restrict__ hist, float* __restrict__ path_out) {
    __shared__ float s[2][Tdim];
    const int j = threadIdx.x;
    if (j < Tdim) s[0][j] = start_t[j] + em[j];
    __syncthreads();
    for (int t = 1; t < SEQ; ++t) {
        const int cur = t & 1, prv = cur ^ 1;
        if (j < Tdim) {
            float best = -3.4e38f; int bi = 0;
            for (int p = 0; p < Tdim; ++p) {
                const float v = s[prv][p] + trans[p * Tdim + j];
                if (v > best) { best = v; bi = p; }
            }
            s[cur][j] = best + em[t * Tdim + j];
            hist[(t - 1) * Tdim + j] = bi;
        }
        __syncthreads();
    }
    if (j == 0) {
        const int fin = (SEQ - 1) & 1;
        float best = -3.4e38f; int last = 0;
        for (int p = 0; p < Tdim; ++p) {
            const float v = s[fin][p] + end_t[p];
            if (v > best) { best = v; last = p; }
        }
        path_out[SEQ - 1] = (float)last;
        for (int t = SEQ - 2; t >= 0; --t) {
            last = hist[t * Tdim + last];
            path_out[t] = (float)last;
        }
    }
}

extern "C" void kernel_launch(void* const* d_in, const int* in_sizes, int n_in,
                              void* d_out, int out_size, void* d_ws, size_t ws_size,
                              hipStream_t stream) {
    const int*   x       = (const int*)d_in[0];
    const int*   casing  = (const int*)d_in[1];
    const int*   pos     = (const int*)d_in[2];
    const float* h0      = (const float*)d_in[3];
    const float* c0      = (const float*)d_in[4];
    const float* W_emb   = (const float*)d_in[5];
    const float* W_cas   = (const float*)d_in[6];
    const float* W_pos   = (const float*)d_in[7];
    const float* Wx_f    = (const float*)d_in[8];
    const float* Wh_f    = (const float*)d_in[9];
    const float* b_f     = (const float*)d_in[10];
    const float* Wx_b    = (const float*)d_in[11];
    const float* Wh_b    = (const float*)d_in[12];
    const float* b_b     = (const float*)d_in[13];
    const float* W_out   = (const float*)d_in[14];
    const float* b_out   = (const float*)d_in[15];
    const float* trans   = (const float*)d_in[16];
    const float* start_t = (const float*)d_in[17];
    const float* end_t   = (const float*)d_in[18];

    float* tag_scores = (float*)d_out;                 // [1024,13]
    float* path_out   = (float*)d_out + SEQ * Tdim;    // [1024]

    // workspace layout
    char* ws = (char*)d_ws;
    float* feat = (float*)ws;                        ws += sizeof(float) * (size_t)SEQ * INDIM;
    float* pre  = (float*)ws;                        ws += sizeof(float) * (size_t)2 * SEQ * H4;
    float* hs   = (float*)ws;                        ws += sizeof(float) * (size_t)2 * SEQ * Hdim;
    float* Hbuf = (float*)ws;                        ws += sizeof(float) * 4 * Hdim;
    int*   hist = (int*)ws;                          ws += sizeof(int) * (size_t)(SEQ - 1) * Tdim;
    int*   cnt  = (int*)ws;                          ws += sizeof(int) * 2;

    hipMemsetAsync(cnt, 0, 2 * sizeof(int), stream);

    gather_kernel<<<SEQ, 128, 0, stream>>>(x, casing, pos, W_emb, W_cas, W_pos, feat);

    preproj_wmma_kernel<<<2048, 256, 0, stream>>>(feat, Wx_f, b_f, Wx_b, b_b, pre);

    const int lstm_smem = (Hdim * 128 + Hdim + 128 + 32) * sizeof(float); // ~258KB (<320KB WGP LDS)
    hipFuncSetAttribute((const void*)lstm_kernel,
                        hipFuncAttributeMaxDynamicSharedMemorySize, lstm_smem);
    lstm_kernel<<<32, 128, lstm_smem, stream>>>(pre, Wh_f, Wh_b, h0, c0, hs, Hbuf, cnt);

    proj_softmax_kernel<<<SEQ, 64, 0, stream>>>(hs, W_out, b_out, tag_scores);

    viterbi_kernel<<<1, 32, 0, stream>>>(tag_scores, trans, start_t, end_t, hist, path_out);
}